// GraphNeuralNetwork_89275190215311
// MI455X (gfx1250) — compile-verified
//
#include <hip/hip_runtime.h>

typedef __attribute__((ext_vector_type(2))) float v2f;
typedef __attribute__((ext_vector_type(4))) float v4f;
typedef __attribute__((ext_vector_type(8))) float v8f;
typedef int v4i_ __attribute__((vector_size(16)));   // matches builtin param type

#define GAS __attribute__((address_space(1)))
#define LAS __attribute__((address_space(3)))

#define DIM     4096
#define MB      256
#define NLAYER  15
#define MTILE   128
#define NTILE   64
#define KC      32
#define AS_STR  36    // A LDS row stride (floats); conflict-free b64 frag reads
#define WS_STR  160   // W LDS dwords per k-pair row; 160 % 64 == 32 -> half-waves disjoint banks

#if __has_builtin(__builtin_amdgcn_global_load_async_to_lds_b128) && \
    __has_builtin(__builtin_amdgcn_global_load_async_to_lds_b32)
#define USE_ASYNC 1
#else
#define USE_ASYNC 0
#endif

#if __has_builtin(__builtin_amdgcn_s_wait_asynccnt)
#define WAIT_ASYNC() __builtin_amdgcn_s_wait_asynccnt(0)
#else
#define WAIT_ASYNC() asm volatile("s_wait_asynccnt 0x0" ::: "memory")
#endif

__global__ __launch_bounds__(256)
void dag_layer_kernel(const float* __restrict__ A,
                      const float* __restrict__ W,
                      const float* __restrict__ bias,
                      float* __restrict__ Out)
{
    __shared__ float As[2][MTILE * AS_STR];       // [m][k]: b64 frag = (k,k+1)
    __shared__ float Ws[2][(KC / 2) * WS_STR];    // pair-interleaved: (k,n) @ (k>>1)*WS_STR + 2n + (k&1)

    const int tid  = threadIdx.x;
    const int lane = tid & 31;
    const int wave = tid >> 5;
    const int wm   = wave & 3;   // M wave offset: wm*32
    const int wn   = wave >> 2;  // N wave offset: wn*32
    const int m0g  = blockIdx.y * MTILE;
    const int n0g  = blockIdx.x * NTILE;

    // A staging map: 128 rows x 32 floats, 16B per lane per issue
    const int a_tr = tid >> 3;        // 0..31
    const int a_tc = (tid & 7) * 4;   // float4 column

#if USE_ASYNC
    auto issue = [&](int k0, int buf) {
#pragma unroll
        for (int rr = 0; rr < 4; ++rr) {
            int row = rr * 32 + a_tr;
            const float* g = A + (size_t)(m0g + row) * DIM + k0 + a_tc;
            float* l = &As[buf][row * AS_STR + a_tc];
            __builtin_amdgcn_global_load_async_to_lds_b128(
                (GAS v4i_*)(GAS void*)g, (LAS v4i_*)(LAS void*)l, 0, 0);
        }
        // W: 32x64 floats, scattering b32 copies into pair-interleaved layout
#pragma unroll
        for (int rr = 0; rr < 8; ++rr) {
            int e = rr * 256 + tid;
            int k = e >> 6, n = e & 63;
            const float* g = W + (size_t)(k0 + k) * DIM + n0g + n;
            float* l = &Ws[buf][(k >> 1) * WS_STR + n * 2 + (k & 1)];
            __builtin_amdgcn_global_load_async_to_lds_b32(
                (GAS int*)(GAS void*)g, (LAS int*)(LAS void*)l, 0, 0);
        }
    };
#else
    const int w_kr = tid >> 4;        // 0..15
    const int w_nc = (tid & 15) * 4;  // float4 column
    v4f pa[4];
    v4f pw[2];
    auto fetch = [&](int k0) {
#pragma unroll
        for (int rr = 0; rr < 4; ++rr) {
            int row = rr * 32 + a_tr;
            pa[rr] = *(const v4f*)(A + (size_t)(m0g + row) * DIM + k0 + a_tc);
        }
#pragma unroll
        for (int rr = 0; rr < 2; ++rr) {
            int k = rr * 16 + w_kr;
            pw[rr] = *(const v4f*)(W + (size_t)(k0 + k) * DIM + n0g + w_nc);
        }
    };
    auto stage = [&](int buf) {
#pragma unroll
        for (int rr = 0; rr < 4; ++rr) {
            int row = rr * 32 + a_tr;
            *(v4f*)(&As[buf][row * AS_STR + a_tc]) = pa[rr];
        }
#pragma unroll
        for (int rr = 0; rr < 2; ++rr) {
            int k = rr * 16 + w_kr;
#pragma unroll
            for (int i = 0; i < 4; ++i)
                Ws[buf][(k >> 1) * WS_STR + (w_nc + i) * 2 + (k & 1)] = pw[rr][i];
        }
    };
#endif

    v8f acc[2][2] = {};

    const int frow  = lane & 15;        // M (A frag) / N (B frag) within subtile
    const int khalf = (lane >> 4) * 2;  // lanes 16-31 hold K+2,K+3

    auto compute = [&](int buf) {
#pragma unroll
        for (int k4 = 0; k4 < KC / 4; ++k4) {
            const int kb = k4 * 4;
            v2f afrag[2], bfrag[2];
#pragma unroll
            for (int i = 0; i < 2; ++i) {
                int m = wm * 32 + i * 16 + frow;
                afrag[i] = *(const v2f*)(&As[buf][m * AS_STR + kb + khalf]);
            }
#pragma unroll
            for (int j = 0; j < 2; ++j) {
                int n = wn * 32 + j * 16 + frow;
                bfrag[j] = *(const v2f*)(&Ws[buf][((kb + khalf) >> 1) * WS_STR + n * 2]);
            }
#pragma unroll
            for (int i = 0; i < 2; ++i)
#pragma unroll
                for (int j = 0; j < 2; ++j)
                    acc[i][j] = __builtin_amdgcn_wmma_f32_16x16x4_f32(
                        false, afrag[i], false, bfrag[j],
                        (short)0, acc[i][j], false, false);
        }
    };

    const int NCHUNK = DIM / KC;

#if USE_ASYNC
    issue(0, 0);
    WAIT_ASYNC();
    __syncthreads();
    for (int kc = 0; kc < NCHUNK; ++kc) {
        const int buf = kc & 1;
        if (kc + 1 < NCHUNK) issue((kc + 1) * KC, buf ^ 1);  // DMA overlaps compute
        compute(buf);
        if (kc + 1 < NCHUNK) {
            WAIT_ASYNC();      // my async copies into buf^1 done
            __syncthreads();   // everyone's done
        }
    }
#else
    fetch(0);
    stage(0);
    __syncthreads();
    for (int kc = 0; kc < NCHUNK; ++kc) {
        const int buf = kc & 1;
        if (kc + 1 < NCHUNK) fetch((kc + 1) * KC);
        compute(buf);
        __syncthreads();
        if (kc + 1 < NCHUNK) {
            stage(buf ^ 1);
            __syncthreads();
        }
    }
#endif

    // epilogue: bias + relu; C frag: VGPR r -> rows r (lanes 0-15) / r+8 (lanes 16-31)
#pragma unroll
    for (int j = 0; j < 2; ++j) {
        int n = n0g + wn * 32 + j * 16 + frow;
        float bv = bias[n];
#pragma unroll
        for (int i = 0; i < 2; ++i) {
            int mbase = m0g + wm * 32 + i * 16 + (lane >> 4) * 8;
#pragma unroll
            for (int r = 0; r < 8; ++r) {
                float v = acc[i][j][r] + bv;
                Out[(size_t)(mbase + r) * DIM + n] = v > 0.f ? v : 0.f;
            }
        }
    }
}

extern "C" void kernel_launch(void* const* d_in, const int* in_sizes, int n_in,
                              void* d_out, int out_size, void* d_ws, size_t ws_size,
                              hipStream_t stream)
{
    const float* x = (const float*)d_in[0];
    const float* W = (const float*)d_in[1];
    const float* b = (const float*)d_in[2];
    float* out = (float*)d_out;
    float* t0  = (float*)d_ws;
    float* t1  = t0 + (size_t)MB * DIM;

    dim3 grid(DIM / NTILE, MB / MTILE);   // 64 x 2 = 128 workgroups, 1024 waves
    dim3 block(256);

    const float* cur = x;
    for (int l = 0; l < NLAYER; ++l) {
        float* dst = (l == NLAYER - 1) ? out : ((l & 1) ? t1 : t0);
        dag_layer_kernel<<<grid, block, 0, stream>>>(
            cur, W + (size_t)l * DIM * DIM, b + (size_t)l * DIM, dst);
        cur = dst;
    }
}